// ExtractGraph_50611894616774
// MI455X (gfx1250) — compile-verified
//
#include <hip/hip_runtime.h>
#include <stdint.h>

#define MM 2048
#define NN 2048
#define PITCH 2064   // padded row pitch (floats) for the intermediate
#define PROWS 2050   // 2048 interior rows + 1 halo row each side

typedef unsigned int v4u __attribute__((ext_vector_type(4)));
typedef int          v8i __attribute__((ext_vector_type(8)));
typedef int          v4i __attribute__((ext_vector_type(4)));

__device__ __forceinline__ unsigned hash_u32(unsigned x) {
    x ^= x >> 16; x *= 0x7feb352dU;
    x ^= x >> 15; x *= 0x846ca68bU;
    x ^= x >> 16;
    return x;
}
__device__ __forceinline__ float u01(unsigned h) {
    return (float)(h >> 8) * (1.0f / 16777216.0f);
}
// monotonic float <-> orderable uint (for integer atomic min/max)
__device__ __forceinline__ unsigned f2ord(float f) {
    unsigned u = __float_as_uint(f);
    return (u & 0x80000000u) ? ~u : (u | 0x80000000u);
}
__device__ __forceinline__ float ord2f(unsigned u) {
    return __uint_as_float((u & 0x80000000u) ? (u ^ 0x80000000u) : ~u);
}

__global__ void init_minmax(unsigned* mm) {
    mm[0] = 0xFFFFFFFFu;  // ordered-min identity
    mm[1] = 0x00000000u;  // ordered-max identity
}

// MaxPool2x2 + uniform noise + global min/max reduction of the pooled map.
__global__ void __launch_bounds__(256) pool_noise_minmax(
    const float* __restrict__ coarse,   // (2*MM) x (2*NN)
    float* __restrict__ noised,         // padded PROWS x PITCH, interior at (+1,+1)
    unsigned* __restrict__ mm)
{
    unsigned gid = blockIdx.x * blockDim.x + threadIdx.x; // MM * (NN/2) threads
    unsigned i  = gid >> 10;        // pooled row (NN/2 = 1024 threads per row)
    unsigned t  = gid & 1023u;
    unsigned j0 = t * 2u;           // first of two pooled cols handled

    const float4* r0 = (const float4*)(coarse + (size_t)(2 * i)     * (2 * NN)) + t;
    const float4* r1 = (const float4*)(coarse + (size_t)(2 * i + 1) * (2 * NN)) + t;
    float4 a = *r0;
    float4 b = *r1;
    float p0 = fmaxf(fmaxf(a.x, a.y), fmaxf(b.x, b.y));
    float p1 = fmaxf(fmaxf(a.z, a.w), fmaxf(b.z, b.w));

    float n0 = u01(hash_u32(i * NN + j0 + 0x9E3779B9u));
    float n1 = u01(hash_u32(i * NN + j0 + 1u + 0x9E3779B9u));

    size_t rowbase = (size_t)(i + 1) * PITCH;
    noised[rowbase + (j0 + 1)] = p0 + n0;
    noised[rowbase + (j0 + 2)] = p1 + n1;

    // threshold uses PRE-noise pooled values
    float lmin = fminf(p0, p1), lmax = fmaxf(p0, p1);
    #pragma unroll
    for (int off = 16; off > 0; off >>= 1) {
        lmin = fminf(lmin, __shfl_xor(lmin, off, 32));
        lmax = fmaxf(lmax, __shfl_xor(lmax, off, 32));
    }
    __shared__ unsigned smin[8], smax[8];
    int lane = threadIdx.x & 31, wave = threadIdx.x >> 5;
    if (lane == 0) { smin[wave] = f2ord(lmin); smax[wave] = f2ord(lmax); }
    __syncthreads();
    if (threadIdx.x == 0) {
        unsigned mn = smin[0], mx = smax[0];
        #pragma unroll
        for (int w = 1; w < 8; ++w) { mn = min(mn, smin[w]); mx = max(mx, smax[w]); }
        atomicMin(&mm[0], mn);
        atomicMax(&mm[1], mx);
    }
}

// Diagonal-neighbor stencil + transpose + dropout. 32x32 output tile per block.
// Halo tile (34x34) is DMA'd into LDS by the Tensor Data Mover.
__global__ void __launch_bounds__(256) recon_graph(
    const float* __restrict__ noised,       // padded PROWS x PITCH
    const unsigned* __restrict__ mm,
    float* __restrict__ out)                // NN x MM (bool as 0.0/1.0 f32)
{
    __shared__ float smem[34 * 34 + 32 * 33];
    float* tile = smem;             // [34][34] halo tile
    float* okf  = smem + 34 * 34;   // [32][33] stencil results (padded vs bank conflicts)

    unsigned bi = blockIdx.y;  // i-tile
    unsigned bj = blockIdx.x;  // j-tile

    if (threadIdx.y == 0) {  // wave-uniform: one wave issues the TDM op
        unsigned long long ga =
            (unsigned long long)(const void*)(noised + (size_t)(bi * 32) * PITCH + bj * 32);
        unsigned lds = (unsigned)(unsigned long long)(void*)tile;  // low 32 bits = LDS offset

        // D# group 0: count=1 (valid), lds_addr, global_addr[56:0], type=2
        v4u g0;
        g0.x = 1u;
        g0.y = lds;
        g0.z = (unsigned)ga;
        g0.w = ((unsigned)(ga >> 32) & 0x01FFFFFFu) | (2u << 30);

        // D# group 1: data_size=4B; tensor dims large (no clipping); tile 34x34;
        // tensor_dim0_stride = PITCH
        const unsigned TD0 = 1u << 20, TD1 = 1u << 20;
        v8i g1;
        g1[0] = (int)(2u << 16);                                        // data_size=2 (4B)
        g1[1] = (int)((TD0 & 0xFFFFu) << 16);                           // tensor_dim0 lo16
        g1[2] = (int)(((TD0 >> 16) & 0xFFFFu) | ((TD1 & 0xFFFFu) << 16));
        g1[3] = (int)(((TD1 >> 16) & 0xFFFFu) | (34u << 16));           // tile_dim0=34
        g1[4] = (int)34u;                                               // tile_dim1=34, tile_dim2=0
        g1[5] = (int)PITCH;                                             // dim0 stride lo32
        g1[6] = 0;
        g1[7] = 0;

        v4i gz4 = {0, 0, 0, 0};              // groups 2/3 unused (2D tensor)
        v8i gz8 = {0, 0, 0, 0, 0, 0, 0, 0};  // trailing group (clang-23 6-arg form)
        __builtin_amdgcn_tensor_load_to_lds(g0, g1, gz4, gz4, gz8, 0);
        __builtin_amdgcn_s_wait_tensorcnt(0);
    }
    __syncthreads();

    float thr = (ord2f(mm[1]) - ord2f(mm[0])) * (1.0f / 2048.0f);

    int lj = threadIdx.x;
    #pragma unroll
    for (int k = 0; k < 4; ++k) {
        int li = threadIdx.y + k * 8;
        int i = (int)(bi * 32) + li;
        int j = (int)(bj * 32) + lj;
        float c = tile[(li + 1) * 34 + (lj + 1)];
        bool any = false;
        #pragma unroll
        for (int dy = -1; dy <= 1; dy += 2) {
            #pragma unroll
            for (int dx = -1; dx <= 1; dx += 2) {
                float nb = tile[(li + 1 + dy) * 34 + (lj + 1 + dx)];
                bool ok = fabsf(nb - c) <= thr;
                bool mask = (i + dx >= 0) & (i + dy < MM) & (j + dx >= 0) &
                            (j + dy < NN) & (i + dy >= 0) & (j + dx < NN);
                any |= (ok & mask);
            }
        }
        okf[li * 33 + lj] = any ? 1.0f : 0.0f;
    }
    __syncthreads();

    // transposed write, coalesced: out[j][i] with i contiguous across lanes
    #pragma unroll
    for (int k = 0; k < 4; ++k) {
        int lj2 = threadIdx.y + k * 8;   // local j of output row
        int li2 = threadIdx.x;           // local i of output col
        int j = (int)(bj * 32) + lj2;
        int i = (int)(bi * 32) + li2;
        float v = okf[li2 * 33 + lj2];
        unsigned keep = hash_u32(((unsigned)(j * NN + i)) ^ 0xB5297A4Du) & 1u;
        out[(size_t)j * NN + i] = (keep && v != 0.0f) ? 1.0f : 0.0f;
    }
}

extern "C" void kernel_launch(void* const* d_in, const int* in_sizes, int n_in,
                              void* d_out, int out_size, void* d_ws, size_t ws_size,
                              hipStream_t stream) {
    const float* coarse = (const float*)d_in[0];
    unsigned char* ws   = (unsigned char*)d_ws;
    unsigned* mm        = (unsigned*)ws;            // 2 cells
    float* noised       = (float*)(ws + 256);       // padded PROWS x PITCH
    float* out          = (float*)d_out;

    init_minmax<<<1, 1, 0, stream>>>(mm);

    int nthreads = MM * (NN / 2);                   // 2 pooled cols per thread
    pool_noise_minmax<<<nthreads / 256, 256, 0, stream>>>(coarse, noised, mm);

    recon_graph<<<dim3(NN / 32, MM / 32), dim3(32, 8), 0, stream>>>(noised, mm, out);
}